// AnomalyDAE_81011673137274
// MI455X (gfx1250) — compile-verified
//
#include <hip/hip_runtime.h>

typedef __attribute__((ext_vector_type(2))) float v2f;
typedef __attribute__((ext_vector_type(8))) float v8f;

#define NN 50000
#define NE 1600000

// ---------------- degree / norm precompute ----------------

__global__ __launch_bounds__(256) void k_init_deg(float* deg) {
  int i = blockIdx.x * 256 + threadIdx.x;
  if (i < NN) deg[i] = 1.0f;  // self-loop contribution
}

__global__ __launch_bounds__(256) void k_deg_edges(const int* __restrict__ dst,
                                                   float* __restrict__ deg) {
  int e = blockIdx.x * 256 + threadIdx.x;
  if (e < NE) {
    __hip_atomic_fetch_add(&deg[dst[e]], 1.0f, __ATOMIC_RELAXED,
                           __HIP_MEMORY_SCOPE_AGENT);
  }
}

__global__ __launch_bounds__(256) void k_finish_dinv(float* deg) {
  int i = blockIdx.x * 256 + threadIdx.x;
  if (i < NN) {
    float d = deg[i];
    d = d < 1e-12f ? 1e-12f : d;
    deg[i] = rsqrtf(d);
  }
}

__global__ __launch_bounds__(256) void k_edge_norm(const int* __restrict__ src,
                                                   const int* __restrict__ dst,
                                                   const float* __restrict__ dinv,
                                                   float* __restrict__ wnorm) {
  int e = blockIdx.x * 256 + threadIdx.x;
  if (e < NE) wnorm[e] = dinv[src[e]] * dinv[dst[e]];
}

// ---------------- WMMA f32 GEMM: H[M,N] = A[M,K] @ B[K,N] ----------------
// One wave32 per 16x16 output tile, V_WMMA_F32_16X16X4_F32, K stepped by 4.
// f32 A 16x4 layout (ISA 7.12.2): lane 0-15 -> K={0,1}, lane 16-31 -> K={2,3};
// B 4x16 mirrors this with N = lane&15. C/D: VGPR r -> M = r + 8*(lane>>4).
__global__ __launch_bounds__(256) void k_gemm_f32_wmma(
    const float* __restrict__ A, const float* __restrict__ B,
    float* __restrict__ H, int M, int K, int N) {
  int wave = (int)(blockIdx.x * 256 + threadIdx.x) >> 5;
  int lane = threadIdx.x & 31;
  int ntiles = N >> 4;
  int mt = wave / ntiles;
  int nt = wave - mt * ntiles;
  if (mt * 16 >= M) return;  // wave-uniform: EXEC stays all-ones for WMMA

  int half = lane >> 4;   // 0 or 1
  int l15  = lane & 15;
  int khalf = half << 1;  // K offset 0 or 2

  const float* ap = A + (size_t)(mt * 16 + l15) * K + khalf;
  const float* bp = B + (size_t)khalf * N + nt * 16 + l15;

  v8f c = {};
  for (int k = 0; k < K; k += 4) {
    float2 av = *(const float2*)ap;  // A[m][khalf], A[m][khalf+1]
    v2f a; a[0] = av.x; a[1] = av.y;
    v2f b; b[0] = bp[0]; b[1] = bp[N];  // B[khalf][n], B[khalf+1][n]
    c = __builtin_amdgcn_wmma_f32_16x16x4_f32(
        /*neg_a=*/false, a, /*neg_b=*/false, b,
        /*c_mod=*/(short)0, c, /*reuse_a=*/false, /*reuse_b=*/false);
    ap += 4;
    bp += (size_t)4 * N;
  }

  float* hp = H + (size_t)(mt * 16 + (half << 3)) * N + nt * 16 + l15;
#pragma unroll
  for (int r = 0; r < 8; ++r) hp[(size_t)r * N] = c[r];
}

// ---------------- aggregation ----------------

// Self-loop term: agg[i,f] = h[i,f] * dinv[i]^2  (full overwrite of agg)
__global__ __launch_bounds__(256) void k_self_init(const float* __restrict__ h,
                                                   const float* __restrict__ dinv,
                                                   float* __restrict__ agg,
                                                   int fshift, long total) {
  long idx = (long)blockIdx.x * 256 + threadIdx.x;
  if (idx >= total) return;
  int node = (int)(idx >> fshift);
  float di = dinv[node];
  agg[idx] = h[idx] * di * di;
}

// Edge scatter: each thread handles one (edge, 4-float chunk).
__global__ __launch_bounds__(256) void k_edge_scatter(
    const int* __restrict__ src, const int* __restrict__ dst,
    const float* __restrict__ wnorm, const float* __restrict__ h,
    float* __restrict__ agg, int F, int cshift, int cmask) {
  long tid = (long)blockIdx.x * 256 + threadIdx.x;
  int e = (int)(tid >> cshift);
  if (e >= NE) return;
  int c = (int)tid & cmask;

  int s = src[e];
  int d = dst[e];
  float w = wnorm[e];
  float4 hv = *((const float4*)(h + (size_t)s * F) + c);
  float* op = agg + (size_t)d * F + ((size_t)c << 2);
  __hip_atomic_fetch_add(op + 0, hv.x * w, __ATOMIC_RELAXED, __HIP_MEMORY_SCOPE_AGENT);
  __hip_atomic_fetch_add(op + 1, hv.y * w, __ATOMIC_RELAXED, __HIP_MEMORY_SCOPE_AGENT);
  __hip_atomic_fetch_add(op + 2, hv.z * w, __ATOMIC_RELAXED, __HIP_MEMORY_SCOPE_AGENT);
  __hip_atomic_fetch_add(op + 3, hv.w * w, __ATOMIC_RELAXED, __HIP_MEMORY_SCOPE_AGENT);
}

// Bias (+ optional ReLU), in place.
__global__ __launch_bounds__(256) void k_epilogue(float* __restrict__ agg,
                                                  const float* __restrict__ bias,
                                                  int fmask, int dorelu, long total) {
  long idx = (long)blockIdx.x * 256 + threadIdx.x;
  if (idx >= total) return;
  float v = agg[idx] + bias[(int)idx & fmask];
  if (dorelu) v = v > 0.0f ? v : 0.0f;
  agg[idx] = v;
}

// ---------------- launcher ----------------

extern "C" void kernel_launch(void* const* d_in, const int* in_sizes, int n_in,
                              void* d_out, int out_size, void* d_ws, size_t ws_size,
                              hipStream_t stream) {
  const float* x  = (const float*)d_in[0];
  const int*   ei = (const int*)d_in[1];
  const float* W[4] = {(const float*)d_in[2], (const float*)d_in[4],
                       (const float*)d_in[6], (const float*)d_in[8]};
  const float* bias[4] = {(const float*)d_in[3], (const float*)d_in[5],
                          (const float*)d_in[7], (const float*)d_in[9]};
  float* out = (float*)d_out;

  char* ws = (char*)d_ws;
  float* dinv   = (float*)(ws);                      // 200 KB
  float* wnorm  = (float*)(ws + (1ull << 20));       // 6.4 MB
  float* hbuf   = (float*)(ws + (8ull << 20));       // 51.2 MB
  float* aggbuf = (float*)(ws + (64ull << 20));      // 25.6 MB

  const int* src = ei;
  const int* dst = ei + NE;

  int nb_nodes = (NN + 255) / 256;
  int nb_edges = (NE + 255) / 256;

  // ---- symmetric-normalization precompute (shared by all 4 layers) ----
  k_init_deg<<<nb_nodes, 256, 0, stream>>>(dinv);
  k_deg_edges<<<nb_edges, 256, 0, stream>>>(dst, dinv);
  k_finish_dinv<<<nb_nodes, 256, 0, stream>>>(dinv);
  k_edge_norm<<<nb_edges, 256, 0, stream>>>(src, dst, dinv, wnorm);

  // ---- one GCN layer ----
  auto layer = [&](const float* in, int K, int li, int F, float* aggdst, int relu) {
    int waves  = (NN / 16) * (F / 16);
    int blocks = (waves + 7) / 8;  // 8 wave32 per 256-thread block
    k_gemm_f32_wmma<<<blocks, 256, 0, stream>>>(in, W[li], hbuf, NN, K, F);

    int fshift = (F == 64) ? 6 : (F == 128) ? 7 : 8;
    long total = (long)NN * F;
    int nbt = (int)((total + 255) / 256);
    k_self_init<<<nbt, 256, 0, stream>>>(hbuf, dinv, aggdst, fshift, total);

    int cshift = fshift - 2;            // log2(F/4) chunks per edge
    int cmask  = (F >> 2) - 1;
    long etot  = (long)NE << cshift;
    int nbe = (int)((etot + 255) / 256);
    k_edge_scatter<<<nbe, 256, 0, stream>>>(src, dst, wnorm, hbuf, aggdst,
                                            F, cshift, cmask);

    k_epilogue<<<nbt, 256, 0, stream>>>(aggdst, bias[li], F - 1, relu, total);
  };

  layer(x,      256, 0, 128, aggbuf, 1);  // L1: 256 -> 128, ReLU
  layer(aggbuf, 128, 1,  64, aggbuf, 1);  // L2: 128 ->  64, ReLU
  layer(aggbuf,  64, 2, 128, aggbuf, 1);  // L3:  64 -> 128, ReLU
  layer(aggbuf, 128, 3, 256, out,    0);  // L4: 128 -> 256, no ReLU -> d_out
}